// FEMloss_4226247819613
// MI455X (gfx1250) — compile-verified
//
#include <hip/hip_runtime.h>
#include <math.h>

// MI455X (gfx1250) notes:
//  - wave32; WMMA F32 16x16x4: A=16x4 (2 VGPR/lane), B=4x16 (2 VGPR/lane), C=16x16 (8 VGPR/lane)
//  - z (64MB) and zT (64MB) both fit in 192MB L2 -> gathers are L2 hits.
//  - Bound by HBM streaming of vals/rows/cols (~108MB) + transpose (~128MB) => ~13us floor.
//  - All hot-loop indexing is int32 (max offset 16M floats) -> GVS addressing
//    (SGPR base + 32-bit voffset), uniform scalar loop via readfirstlane.

typedef __attribute__((ext_vector_type(2))) float v2f;
typedef __attribute__((ext_vector_type(8))) float v8f;

#define WAVE 32
#define TB   256   // 8 waves / block
#define NBLK 2048

__global__ void zero_quad_kernel(float* __restrict__ quad) {
    if (threadIdx.x < 16) quad[threadIdx.x] = 0.0f;
}

// z is (16, N) row-major; produce zT (N, 16) so each node's 16-batch vector is
// one contiguous 64B line (single cache line per gather instead of 16).
__global__ void transpose_z_kernel(const float* __restrict__ z,
                                   float* __restrict__ zT, int N) {
    int i = blockIdx.x * blockDim.x + threadIdx.x;
    if (i >= N) return;
    float4 out[4];
#pragma unroll
    for (int q = 0; q < 4; ++q) {
        out[q].x = z[(long long)(4 * q + 0) * N + i];   // coalesced across lanes
        out[q].y = z[(long long)(4 * q + 1) * N + i];
        out[q].z = z[(long long)(4 * q + 2) * N + i];
        out[q].w = z[(long long)(4 * q + 3) * N + i];
    }
    float4* dst = (float4*)(zT + (long long)i * 16);    // 64B contiguous store per lane
#pragma unroll
    for (int q = 0; q < 4; ++q) dst[q] = out[q];
}

// Each wave consumes groups of 4 nonzeros per V_WMMA_F32_16X16X4_F32:
//   A[b][k] = vals[k] * zv[rows[k]][b]   (16x4)
//   B[k][b] = zv[cols[k]][b]             (4x16)
//   C[b][b] accumulates quad[b]; off-diagonal discarded (memory-bound anyway).
// Element (node, batch) at zv[node*sNode + batch*sB]:
//   transposed path: sNode=16, sB=1 ; fallback (no ws): sNode=1, sB=N.
__global__ void __launch_bounds__(TB)
quad_wmma_kernel(const float* __restrict__ zv,
                 const float* __restrict__ vals,
                 const int* __restrict__ rows,
                 const int* __restrict__ cols,
                 float* __restrict__ quad,
                 int nnz, int sNode, int sB) {
    __shared__ float lds[TB / WAVE][8][WAVE];

    const int lane = threadIdx.x & (WAVE - 1);
    const int w    = threadIdx.x >> 5;
    const int m    = lane & 15;          // matrix row (A) / col (B) this lane feeds
    const bool hi  = (lane >= 16);       // lanes 0-15 -> k {0,1}; 16-31 -> k {2,3}

    // Uniform (SGPR) wave id -> scalar loop, EXEC stays all-1s (WMMA requirement).
    const int waveId = __builtin_amdgcn_readfirstlane(
        blockIdx.x * (TB / WAVE) + (threadIdx.x >> 5));
    const int nWaves  = gridDim.x * (TB / WAVE);
    const int nGroups = (nnz + 3) >> 2;

    v8f c = {};  // 16x16 f32 accumulator (diagonal = quad[b])

    for (int g = waveId; g < nGroups; g += nWaves) {
        // Branchless tail: shift a partial last group down to nnz-4 and zero
        // the weights that would double-count. All conditions are uniform.
        const int base  = g << 2;
        const int baseC = (base + 4 <= nnz) ? base : (nnz - 4);
        const int skip  = base - baseC;            // 0 except in the tail group

        // Uniform, 16B-aligned index/value loads -> s_load_b128 via K$.
        float v0 = vals[baseC + 0], v1 = vals[baseC + 1];
        float v2 = vals[baseC + 2], v3 = vals[baseC + 3];
        const int r0 = rows[baseC + 0], r1 = rows[baseC + 1];
        const int r2 = rows[baseC + 2], r3 = rows[baseC + 3];
        const int c0 = cols[baseC + 0], c1 = cols[baseC + 1];
        const int c2 = cols[baseC + 2], c3 = cols[baseC + 3];
        if (skip > 0) v0 = 0.0f;
        if (skip > 1) v1 = 0.0f;
        if (skip > 2) v2 = 0.0f;

        // Prefetch our next group's streaming data (global_prefetch_b8;
        // speculative OOB prefetches are silently dropped).
        const int nxt = baseC + (nWaves << 2);
        __builtin_prefetch(vals + nxt, 0, 0);
        __builtin_prefetch(rows + nxt, 0, 0);
        __builtin_prefetch(cols + nxt, 0, 0);

        // Per-lane k selection from scalar regs (v_cndmask), then 4 gathers.
        const float vA0 = hi ? v2 : v0;
        const float vA1 = hi ? v3 : v1;
        const int   rA0 = hi ? r2 : r0;
        const int   rA1 = hi ? r3 : r1;
        const int   cA0 = hi ? c2 : c0;
        const int   cA1 = hi ? c3 : c1;

        const int mOff = m * sB;
        v2f a, b;
        a.x = vA0 * zv[rA0 * sNode + mOff];   // A[m][k0]
        a.y = vA1 * zv[rA1 * sNode + mOff];   // A[m][k1]
        b.x = zv[cA0 * sNode + mOff];         // B[k0][m]
        b.y = zv[cA1 * sNode + mOff];         // B[k1][m]

        // 8 args: (neg_a, A, neg_b, B, c_mod, C, reuse_a, reuse_b)
        c = __builtin_amdgcn_wmma_f32_16x16x4_f32(
                false, a, false, b, (short)0, c, false, false);
    }

    // Extract diagonal C[b][b] via LDS.
    // C layout: VGPR j -> lanes 0-15 hold (M=j, N=lane); lanes 16-31 (M=j+8, N=lane-16).
    // Diagonal b lives in VGPR (b&7), lane (b<8 ? b : b+16).
#pragma unroll
    for (int j = 0; j < 8; ++j) lds[w][j][lane] = c[j];
    __syncthreads();
    if (lane < 16) {
        const int b2 = lane;
        const int srcLane = (b2 < 8) ? b2 : (b2 + 16);
        atomicAdd(&quad[b2], lds[w][b2 & 7][srcLane]);
    }
}

__global__ void finalize_kernel(const float* __restrict__ quad,
                                float* __restrict__ out) {
    if (threadIdx.x == 0) {
        float s = 0.0f;
#pragma unroll
        for (int b = 0; b < 16; ++b) s += sqrtf(quad[b]);
        out[0] = s * (1.0f / 16.0f);
    }
}

extern "C" void kernel_launch(void* const* d_in, const int* in_sizes, int n_in,
                              void* d_out, int out_size, void* d_ws, size_t ws_size,
                              hipStream_t stream) {
    const float* z    = (const float*)d_in[0];   // (16, N) f32
    const float* vals = (const float*)d_in[1];   // (nnz,)  f32
    const int*   rows = (const int*)d_in[2];     // (nnz,)  int
    const int*   cols = (const int*)d_in[3];     // (nnz,)  int

    const int nnz = in_sizes[1];
    const int N   = in_sizes[0] / 16;

    float* quad = (float*)d_ws;                          // 16 f32 accumulators
    float* zT   = (float*)((char*)d_ws + 256);           // (N,16) f32, 256B-aligned
    const size_t need = 256 + (size_t)N * 16 * sizeof(float);

    zero_quad_kernel<<<1, 32, 0, stream>>>(quad);

    const float* zv; int sNode, sB;
    if (ws_size >= need) {
        const int tb = 256;
        transpose_z_kernel<<<(N + tb - 1) / tb, tb, 0, stream>>>(z, zT, N);
        zv = zT; sNode = 16; sB = 1;      // contiguous 64B per gathered node
    } else {
        zv = z;  sNode = 1;  sB = N;      // strided fallback, no scratch needed
    }

    quad_wmma_kernel<<<NBLK, TB, 0, stream>>>(zv, vals, rows, cols, quad,
                                              nnz, sNode, sB);
    finalize_kernel<<<1, 32, 0, stream>>>(quad, (float*)d_out);
}